// BinarySegmentationLoss_47090021433737
// MI455X (gfx1250) — compile-verified
//
#include <hip/hip_runtime.h>

// ---------------------------------------------------------------------------
// BinarySegmentationLoss for MI455X (gfx1250, wave32).
// 403 MB streamed once -> bandwidth-bound (~17us @ 23.3 TB/s).
// Strategy: 128-bit NT global loads, register accumulation of the 10 needed
// statistics, block reduction on the matrix pipe (V_WMMA_F32_16X16X4_F32 with
// ones A-matrix => exact f32 cross-lane sums), then f32 global atomics.
// ---------------------------------------------------------------------------

typedef __attribute__((ext_vector_type(2))) float v2f;
typedef __attribute__((ext_vector_type(4))) float v4f;
typedef __attribute__((ext_vector_type(8))) float v8f;

#define BQ 16
#define CQ 3
#define HWQ (1024 * 1024)
#define GROUPS (HWQ / 4)          // float4 pixel groups per channel plane
#define BLOCKS_PER_IMG 128
#define THREADS 256
#define ACC_STRIDE 16             // 10 used accumulators, padded to 16

// huber(e) with delta=1:  a=|e|, m=min(a,1), h = 0.5*m*(2a - m)
//   a<1 : 0.5*a*(2a-a) = 0.5 a^2 ; a>=1 : 0.5*(2a-1) = a-0.5
__device__ __forceinline__ float huber_f(float e) {
    float a = __builtin_fabsf(e);
    float m = fminf(a, 1.0f);
    return 0.5f * m * (2.0f * a - m);
}

__global__ void zero_ws_kernel(float* __restrict__ ws) {
    int i = threadIdx.x;
    if (i < BQ * ACC_STRIDE) ws[i] = 0.0f;
}

__global__ __launch_bounds__(THREADS)
void reduce_kernel(const float* __restrict__ pred,
                   const float* __restrict__ tgt,
                   float* __restrict__ ws) {
    const int b   = blockIdx.x / BLOCKS_PER_IMG;
    const int blk = blockIdx.x % BLOCKS_PER_IMG;
    const int tid = threadIdx.x;

    const size_t base = (size_t)b * CQ * GROUPS;
    const v4f* p0 = (const v4f*)pred + base;
    const v4f* p1 = p0 + GROUPS;
    const v4f* p2 = p1 + GROUPS;
    const v4f* t0 = (const v4f*)tgt + base;
    const v4f* t1 = t0 + GROUPS;
    const v4f* t2 = t1 + GROUPS;

    float nbg = 0.f, nfg = 0.f, hbg = 0.f, hfg = 0.f;
    float sb0 = 0.f, sb1 = 0.f, sb2 = 0.f;
    float sf0 = 0.f, sf1 = 0.f, sf2 = 0.f;

    for (int g = blk * THREADS + tid; g < GROUPS; g += BLOCKS_PER_IMG * THREADS) {
        // six global_load_b128 (non-temporal: streamed once, keep L2 clean)
        v4f ta = __builtin_nontemporal_load(t0 + g);
        v4f tb = __builtin_nontemporal_load(t1 + g);
        v4f tc = __builtin_nontemporal_load(t2 + g);
        v4f pa = __builtin_nontemporal_load(p0 + g);
        v4f pb = __builtin_nontemporal_load(p1 + g);
        v4f pc = __builtin_nontemporal_load(p2 + g);
#pragma unroll
        for (int i = 0; i < 4; ++i) {
            float tai = ta[i], tbi = tb[i], tci = tc[i];
            float pai = pa[i], pbi = pb[i], pci = pc[i];
            float mbg = (tai == 0.0f   && tbi == 0.0f   && tci == 0.0f)   ? 1.0f : 0.0f;
            float mfg = (tai == 255.0f && tbi == 255.0f && tci == 255.0f) ? 1.0f : 0.0f;
            nbg += mbg;
            nfg += mfg;
            float hb = huber_f(pai) + huber_f(pbi) + huber_f(pci);
            float hf = huber_f(pai - 255.0f) + huber_f(pbi - 255.0f) + huber_f(pci - 255.0f);
            hbg = fmaf(mbg, hb, hbg);
            hfg = fmaf(mfg, hf, hfg);
            sb0 = fmaf(mbg, pai, sb0);
            sb1 = fmaf(mbg, pbi, sb1);
            sb2 = fmaf(mbg, pci, sb2);
            sf0 = fmaf(mfg, pai, sf0);
            sf1 = fmaf(mfg, pbi, sf1);
            sf2 = fmaf(mfg, pci, sf2);
        }
    }

    // ---- block reduction: 256 lanes x 16 accumulator columns ----
    __shared__ float P[THREADS][ACC_STRIDE];
    P[tid][0] = nbg; P[tid][1] = nfg; P[tid][2] = hbg; P[tid][3] = hfg;
    P[tid][4] = sb0; P[tid][5] = sb1; P[tid][6] = sb2;
    P[tid][7] = sf0; P[tid][8] = sf1; P[tid][9] = sf2;
#pragma unroll
    for (int j = 10; j < ACC_STRIDE; ++j) P[tid][j] = 0.0f;
    __syncthreads();

    // Wave 0 (full EXEC, as WMMA requires) sums the 256x16 matrix column-wise
    // on the matrix pipe: D += Ones(16x4) x B(4x16), chained 64 times.
    // Only the N coordinate (= lane & 15) of each B slot matters for a pure
    // K-sum, so the K-slot permutation across lanes/VGPRs is harmless.
    if (tid < 32) {
        const int n    = tid & 15;
        const int half = tid >> 4;     // 0 or 1
        v2f onesm;
        onesm[0] = 1.0f;
        onesm[1] = 1.0f;
        v8f d = {};
        for (int k = 0; k < THREADS; k += 4) {
            v2f bm;
            bm[0] = P[k + half][n];
            bm[1] = P[k + 2 + half][n];
            // (neg_a, A, neg_b, B, c_mod, C, reuse_a, reuse_b)
            d = __builtin_amdgcn_wmma_f32_16x16x4_f32(false, onesm, false, bm,
                                                      (short)0, d, false, false);
        }
        // every row of D is identical; D vgpr0, lanes 0..15 hold column totals
        if (tid < 10) atomicAdd(&ws[b * ACC_STRIDE + tid], d[0]);
    }
}

__global__ void finalize_kernel(const float* __restrict__ ws, float* __restrict__ out) {
    __shared__ float ls[BQ];
    int b = threadIdx.x;
    if (b < BQ) {
        const float* a = ws + b * ACC_STRIDE;
        float nbg = a[0], nfg = a[1], hbg = a[2], hfg = a[3];
        float sb0 = a[4], sb1 = a[5], sb2 = a[6];
        float sf0 = a[7], sf1 = a[8], sf2 = a[9];
        bool has_bg = nbg > 0.0f;
        bool has_fg = nfg > 0.0f;
        bool both   = has_bg && has_fg;
        float safe_bg = fmaxf(nbg, 1.0f);
        float safe_fg = fmaxf(nfg, 1.0f);
        float loss_bg = hbg / (safe_bg * (float)CQ);
        float loss_fg = hfg / (safe_fg * (float)CQ);
        float d0 = sb0 / safe_bg - sf0 / safe_fg;
        float d1 = sb1 / safe_bg - sf1 / safe_fg;
        float d2 = sb2 / safe_bg - sf2 / safe_fg;
        float dist = d0 * d0 + d1 * d1 + d2 * d2;
        float sep = 300.0f / (1.0f + dist);
        float valid = (has_bg ? 1.0f : 0.0f) + (has_fg ? 1.0f : 0.0f) + (both ? 1.0f : 0.0f);
        float loss  = (has_bg ? loss_bg : 0.0f) + (has_fg ? loss_fg : 0.0f) + (both ? sep : 0.0f);
        ls[b] = (valid > 0.0f) ? (loss / valid) : 0.0f;
    }
    __syncthreads();
    if (b == 0) {
        float s = 0.0f;
        for (int i = 0; i < BQ; ++i) s += ls[i];
        *out = s / (float)BQ;
    }
}

extern "C" void kernel_launch(void* const* d_in, const int* in_sizes, int n_in,
                              void* d_out, int out_size, void* d_ws, size_t ws_size,
                              hipStream_t stream) {
    const float* pred = (const float*)d_in[0];
    const float* tgt  = (const float*)d_in[1];
    float* ws  = (float*)d_ws;
    float* out = (float*)d_out;

    zero_ws_kernel<<<1, 256, 0, stream>>>(ws);
    reduce_kernel<<<BQ * BLOCKS_PER_IMG, THREADS, 0, stream>>>(pred, tgt, ws);
    finalize_kernel<<<1, 32, 0, stream>>>(ws, out);
}